// SpecAugment_34720515621664
// MI455X (gfx1250) — compile-verified
//
#include <hip/hip_runtime.h>
#include <stdint.h>

// ---------------------------------------------------------------------------
// SpecAugment for x:(64,1,4000,128) fp32. Memory-bound: 262 MB traffic,
// ~11us floor at 23.3 TB/s HBM. Mask is derived on-device from JAX
// threefry2x32 (key(1)) band parameters: 48 bytes per batch.
// ---------------------------------------------------------------------------

#define B_   64
#define T_   4000
#define F_   128
#define TW_  35      // TIME_MASK width
#define FW_  15      // FREQ_MASK width
#define ROWS_PER_BLOCK 32
#define BLOCKS_PER_B   (T_ / ROWS_PER_BLOCK)   // 125

typedef __attribute__((ext_vector_type(4))) float v4f;
typedef __attribute__((ext_vector_type(4))) int   v4i;

// -------------------- gfx1250 async global->LDS path -----------------------
#if defined(__gfx1250__) && __has_builtin(__builtin_amdgcn_global_load_async_to_lds_b128)
#define HAVE_ASYNC_LDS 1
#else
#define HAVE_ASYNC_LDS 0
#endif

// -------------------- JAX threefry2x32 (exact) ------------------------------
__device__ __forceinline__ uint32_t rotl32(uint32_t v, int d) {
  return (v << d) | (v >> (32 - d));
}

__device__ __forceinline__ void threefry2x32(uint32_t k0, uint32_t k1,
                                             uint32_t c0, uint32_t c1,
                                             uint32_t &o0, uint32_t &o1) {
  const uint32_t ks2 = k0 ^ k1 ^ 0x1BD11BDAu;
  uint32_t x0 = c0 + k0, x1 = c1 + k1;
#define TF_R(r) { x0 += x1; x1 = rotl32(x1, (r)); x1 ^= x0; }
  TF_R(13) TF_R(15) TF_R(26) TF_R(6)  x0 += k1;  x1 += ks2 + 1u;
  TF_R(17) TF_R(29) TF_R(16) TF_R(24) x0 += ks2; x1 += k0  + 2u;
  TF_R(13) TF_R(15) TF_R(26) TF_R(6)  x0 += k0;  x1 += k1  + 3u;
  TF_R(17) TF_R(29) TF_R(16) TF_R(24) x0 += k1;  x1 += ks2 + 4u;
  TF_R(13) TF_R(15) TF_R(26) TF_R(6)  x0 += ks2; x1 += k0  + 5u;
#undef TF_R
  o0 = x0; o1 = x1;
}

// JAX uniform [0,1): bitcast((bits>>9)|0x3f800000) - 1, then max(0, .)
__device__ __forceinline__ float bits_to_u01(uint32_t bits) {
  const uint32_t v = (bits >> 9) | 0x3f800000u;
  const float f = __uint_as_float(v) - 1.0f;
  return f < 0.0f ? 0.0f : f;
}

// ---------------------------------------------------------------------------
// Setup: reproduce
//   kt_a, kt_s, kf_a, kf_s = split(key(1), 4)    (counts iota(8))
//   uniform(k, (64,3)) for each                  (counts iota(192))
// params[b*12 + {0..2}] = t_apply, {3..5} = t_start,
//              {6..8}   = f_apply, {9..11}= f_start
// ---------------------------------------------------------------------------
__global__ void __launch_bounds__(192) spec_setup(int* __restrict__ params) {
  const int j = threadIdx.x;            // 0..191 == flat (b,p) index
  if (j >= B_ * 3) return;

  // split(key(1),4): key data = (0,1); lanewise cipher of (i, i+4), i=0..3.
  // flat out = [A0..A3, B0..B3] -> subkeys rows (A0,A1),(A2,A3),(B0,B1),(B2,B3)
  uint32_t A[4], B[4];
#pragma unroll
  for (int i = 0; i < 4; ++i)
    threefry2x32(0u, 1u, (uint32_t)i, (uint32_t)(i + 4), A[i], B[i]);
  const uint32_t sk[4][2] = {{A[0], A[1]},   // kt_a
                             {A[2], A[3]},   // kt_s
                             {B[0], B[1]},   // kf_a
                             {B[2], B[3]}};  // kf_s

  // uniform(key, (64,3)): size 192; lane i ciphers (i, i+96);
  // flat bits = [a_0..a_95, b_0..b_95]; element j picks word (j<96 ? a_j : b_{j-96})
  const uint32_t i    = (j < 96) ? (uint32_t)j : (uint32_t)(j - 96);
  const int      word = (j < 96) ? 0 : 1;

  float u[4];
#pragma unroll
  for (int s = 0; s < 4; ++s) {
    uint32_t o0, o1;
    threefry2x32(sk[s][0], sk[s][1], i, i + 96u, o0, o1);
    u[s] = bits_to_u01(word ? o1 : o0);
  }

  const int b = j / 3, p = j % 3;
  int* pb = params + b * 12;
  pb[0 + p] = (u[0] < 0.5f) ? 1 : 0;                    // t apply (rate 0.5)
  pb[3 + p] = (int)floorf(u[1] * (float)(T_ - TW_));    // t start, *3965
  pb[6 + p] = (u[2] < 0.5f) ? 1 : 0;                    // f apply
  pb[9 + p] = (int)floorf(u[3] * (float)(F_ - FW_));    // f start, *113
}

// ---------------------------------------------------------------------------
// Main streaming kernel. One block = one batch slice of 32 time rows.
// 8 waves/block; each wave32 owns a 128-float row (32 lanes x float4) and
// processes 4 rows. Params staged via gfx1250 async global->LDS copy.
// ---------------------------------------------------------------------------
__global__ void __launch_bounds__(256)
spec_main(const float* __restrict__ x, float* __restrict__ out,
          const int* __restrict__ params) {
  __shared__ int sp[16];
  const int tid  = threadIdx.x;
  const int b    = blockIdx.x / BLOCKS_PER_B;
  const int blkT = blockIdx.x % BLOCKS_PER_B;

#if HAVE_ASYNC_LDS
  if (tid < 3) {
    __builtin_amdgcn_global_load_async_to_lds_b128(
        (__attribute__((address_space(1))) v4i*)(params + b * 12 + tid * 4),
        (__attribute__((address_space(3))) v4i*)(&sp[tid * 4]),
        0, 0);
  }
#if __has_builtin(__builtin_amdgcn_s_wait_asynccnt)
  __builtin_amdgcn_s_wait_asynccnt(0);
#else
  asm volatile("s_wait_asynccnt 0" ::: "memory");
#endif
#else
  if (tid < 12) sp[tid] = params[b * 12 + tid];
#endif
  __syncthreads();

  const int wave = tid >> 5;   // 0..7
  const int lane = tid & 31;   // 0..31

  // Per-lane frequency keep mask for f = 4*lane .. 4*lane+3 (reused 4 rows).
  float m0, m1, m2, m3;
  {
    float mm[4];
#pragma unroll
    for (int q = 0; q < 4; ++q) {
      const int f = lane * 4 + q;
      int keep = 1;
#pragma unroll
      for (int p = 0; p < 3; ++p) {
        const int a = sp[6 + p], s = sp[9 + p];
        if (a && f >= s && f < s + FW_) keep = 0;
      }
      mm[q] = keep ? 1.0f : 0.0f;
    }
    m0 = mm[0]; m1 = mm[1]; m2 = mm[2]; m3 = mm[3];
  }
  const v4f kf   = (v4f){m0, m1, m2, m3};
  const v4f zero = (v4f)0.0f;

#pragma unroll
  for (int i = 0; i < 4; ++i) {
    const int t = blkT * ROWS_PER_BLOCK + wave * 4 + i;
    int keep_t = 1;
#pragma unroll
    for (int p = 0; p < 3; ++p) {
      const int a = sp[p], s = sp[3 + p];
      if (a && t >= s && t < s + TW_) keep_t = 0;
    }
    const v4f scale = keep_t ? kf : zero;

    const size_t base = ((size_t)b * T_ + (size_t)t) * F_;
    const v4f* __restrict__ xin = (const v4f*)(x + base);
    v4f* __restrict__ xo        = (v4f*)(out + base);

    // Pure streaming: non-temporal b128 load/store (don't pollute L2/WGP$).
    v4f v = __builtin_nontemporal_load(xin + lane);
    __builtin_nontemporal_store(v * scale, xo + lane);
  }
}

// ---------------------------------------------------------------------------
extern "C" void kernel_launch(void* const* d_in, const int* in_sizes, int n_in,
                              void* d_out, int out_size, void* d_ws, size_t ws_size,
                              hipStream_t stream) {
  (void)in_sizes; (void)n_in; (void)out_size; (void)ws_size;
  const float* x = (const float*)d_in[0];
  float* out     = (float*)d_out;
  int* params    = (int*)d_ws;   // 64 * 12 ints = 3 KB

  spec_setup<<<dim3(1), dim3(192), 0, stream>>>(params);
  spec_main<<<dim3(B_ * BLOCKS_PER_B), dim3(256), 0, stream>>>(x, out, params);
}